// Penetration_D_18253611008153
// MI455X (gfx1250) — compile-verified
//
#include <hip/hip_runtime.h>
#include <math.h>

typedef float v2f __attribute__((ext_vector_type(2)));
typedef float v8f __attribute__((ext_vector_type(8)));
typedef int   v4i __attribute__((ext_vector_type(4)));

#define BN_EPS 1e-5f

#define GLOBAL_AS __attribute__((address_space(1)))
#define LDS_AS    __attribute__((address_space(3)))

#if defined(__has_builtin)
#if __has_builtin(__builtin_amdgcn_global_load_async_to_lds_b128)
#define HAVE_ASYNC_LDS 1
#endif
#endif

// ---------------------------------------------------------------- conv1
// body_mesh [32,20813,3] -> x1 [32,3,3401] ; stride 6 along rows, VALID
__global__ void conv1_kernel(const float* __restrict__ mesh,
                             const float* __restrict__ w,      // [3,1,409,3]
                             const float* __restrict__ bias,   // [3]
                             float* __restrict__ x1)
{
    __shared__ float sw[3 * 409 * 3];
    for (int i = threadIdx.x; i < 3 * 409 * 3; i += blockDim.x) sw[i] = w[i];
    __syncthreads();
    int gid = blockIdx.x * blockDim.x + threadIdx.x;
    const int total = 32 * 3 * 3401;
    if (gid >= total) return;
    int h = gid % 3401;
    int c = (gid / 3401) % 3;
    int b = gid / (3 * 3401);
    const float* in = mesh + (size_t)b * 20813 * 3 + (size_t)h * 6 * 3;
    const float* wc = sw + c * 409 * 3;
    float acc = bias[c];
    #pragma unroll 4
    for (int r = 0; r < 409; ++r) {
        acc += in[r*3+0]*wc[r*3+0] + in[r*3+1]*wc[r*3+1] + in[r*3+2]*wc[r*3+2];
    }
    x1[gid] = acc;
}

// ------------------------------------------------- batchnorm2d statistics
// X [B,C,L]; per channel: scale = g*rsqrt(var+eps), shift = beta - mean*scale
__global__ void bn2d_stats_kernel(const float* __restrict__ X,
                                  const float* __restrict__ g,
                                  const float* __restrict__ beta,
                                  float* __restrict__ scale,
                                  float* __restrict__ shift,
                                  int C, int L, int Bn)
{
    __shared__ float rs[256], rs2[256];
    int c = blockIdx.x;
    int n = Bn * L;
    float s = 0.f, s2 = 0.f;
    for (int i = threadIdx.x; i < n; i += blockDim.x) {
        int b = i / L, p = i - b * L;
        float v = X[((size_t)b * C + c) * L + p];
        s += v; s2 += v * v;
    }
    rs[threadIdx.x] = s; rs2[threadIdx.x] = s2;
    __syncthreads();
    for (int off = 128; off > 0; off >>= 1) {
        if (threadIdx.x < off) {
            rs[threadIdx.x]  += rs[threadIdx.x + off];
            rs2[threadIdx.x] += rs2[threadIdx.x + off];
        }
        __syncthreads();
    }
    if (threadIdx.x == 0) {
        float mean = rs[0] / (float)n;
        float var  = rs2[0] / (float)n - mean * mean;
        float sc   = g[c] * rsqrtf(var + BN_EPS);
        scale[c] = sc;
        shift[c] = beta[c] - mean * sc;
    }
}

// ---------------------------------------------------------------- conv2
// reads x1 [32,3,3401] with bn1 applied on the fly, logical layout [B,1,3401,3]
// -> x2 [32,3,3072]
__global__ void conv2_kernel(const float* __restrict__ x1,
                             const float* __restrict__ sc,
                             const float* __restrict__ sh,
                             const float* __restrict__ w,      // [3,1,330,3]
                             const float* __restrict__ bias,
                             float* __restrict__ x2)
{
    __shared__ float sw[3 * 330 * 3];
    __shared__ float s_sc[3], s_sh[3];
    for (int i = threadIdx.x; i < 3 * 330 * 3; i += blockDim.x) sw[i] = w[i];
    if (threadIdx.x < 3) { s_sc[threadIdx.x] = sc[threadIdx.x]; s_sh[threadIdx.x] = sh[threadIdx.x]; }
    __syncthreads();
    int gid = blockIdx.x * blockDim.x + threadIdx.x;
    const int total = 32 * 3 * 3072;
    if (gid >= total) return;
    int oh = gid % 3072;
    int c  = (gid / 3072) % 3;
    int b  = gid / (3 * 3072);
    const float* xb = x1 + (size_t)b * 10203;
    const float* wc = sw + c * 330 * 3;
    float acc = bias[c];
    for (int r = 0; r < 330; ++r) {
        int base = (oh + r) * 3;
        #pragma unroll
        for (int ww = 0; ww < 3; ++ww) {
            int flat = base + ww;            // index into [3401,3] view
            int c1   = flat / 3401;          // back to [3,3401] storage
            int pos  = flat - c1 * 3401;
            float v  = xb[c1 * 3401 + pos] * s_sc[c1] + s_sh[c1];
            acc += v * wc[r*3 + ww];
        }
    }
    x2[gid] = acc;
}

// ----------------------------------------- bn2 apply + PReLU (elementwise)
__global__ void bn2d_apply_prelu_kernel(const float* __restrict__ x,
                                        const float* __restrict__ sc,
                                        const float* __restrict__ sh,
                                        const float* __restrict__ alpha,
                                        float* __restrict__ y)
{
    int i = blockIdx.x * blockDim.x + threadIdx.x;
    if (i >= 32 * 9216) return;
    int c = (i % 9216) / 3072;
    float v = x[i] * sc[c] + sh[c];
    float a = *alpha;
    y[i] = v >= 0.f ? v : a * v;
}

// ----------------------------------------------------------- QKV project
// xp [32, 9216] viewed as [32,1024,9]; out q/k/v as [b][h][f][d]
__global__ void qkv_kernel(const float* __restrict__ xp,
                           const float* __restrict__ wq, const float* __restrict__ bq,
                           const float* __restrict__ wk, const float* __restrict__ bk,
                           const float* __restrict__ wv, const float* __restrict__ bv,
                           float* __restrict__ qb, float* __restrict__ kb,
                           float* __restrict__ vb)
{
    int gid = blockIdx.x * blockDim.x + threadIdx.x;
    if (gid >= 32 * 1024) return;
    int b = gid >> 10, f = gid & 1023;
    const float* xr = xp + (size_t)b * 9216 + f * 9;
    float xin[9];
    #pragma unroll
    for (int j = 0; j < 9; ++j) xin[j] = xr[j];
    #pragma unroll
    for (int c = 0; c < 4; ++c) {
        float aq = bq[c], ak = bk[c], av = bv[c];
        #pragma unroll
        for (int j = 0; j < 9; ++j) {
            aq += wq[c*9+j] * xin[j];
            ak += wk[c*9+j] * xin[j];
            av += wv[c*9+j] * xin[j];
        }
        int h = c >> 1, d = c & 1;
        size_t o = (((size_t)b * 2 + h) * 1024 + f) * 2 + d;
        qb[o] = aq; kb[o] = ak; vb[o] = av;
    }
}

// ------------------------------------- fused attention (never materialize P)
// one block per (b,h); K,V staged into LDS with the CDNA5 async copy engine
// (GLOBAL_LOAD_ASYNC_TO_LDS_B128, tracked by ASYNCcnt), 16 KB of 320 KB WGP LDS
__global__ void attn_kernel(const float* __restrict__ qb,
                            const float* __restrict__ kb,
                            const float* __restrict__ vb,
                            float* __restrict__ avbuf)   // [32,4096] (b, f*4+h*2+d)
{
    __shared__ __align__(16) float sK[2048];
    __shared__ __align__(16) float sV[2048];
    int bh = blockIdx.x;
    int b = bh >> 1, h = bh & 1;
    size_t base = (size_t)bh * 2048;
#if defined(HAVE_ASYNC_LDS)
    // each lane async-copies 16B (one v4i); 256 threads x 2 iters cover 2048 floats
    for (int i = threadIdx.x * 4; i < 2048; i += blockDim.x * 4) {
        __builtin_amdgcn_global_load_async_to_lds_b128(
            (GLOBAL_AS v4i*)(kb + base + i), (LDS_AS v4i*)&sK[i], 0, 0);
        __builtin_amdgcn_global_load_async_to_lds_b128(
            (GLOBAL_AS v4i*)(vb + base + i), (LDS_AS v4i*)&sV[i], 0, 0);
    }
    asm volatile("s_wait_asynccnt 0x0" ::: "memory");
#else
    for (int i = threadIdx.x; i < 2048; i += blockDim.x) {
        sK[i] = kb[base + i];
        sV[i] = vb[base + i];
    }
#endif
    __syncthreads();
    const float rs = 0.70710678118654752440f;   // 1/sqrt(head_dim=2)
    for (int r = 0; r < 4; ++r) {
        int f = threadIdx.x + 256 * r;
        float q0 = qb[base + f*2], q1 = qb[base + f*2 + 1];
        float mx = -1e30f;
        for (int k = 0; k < 1024; ++k) {
            float s = (q0 * sK[2*k] + q1 * sK[2*k+1]) * rs;
            mx = fmaxf(mx, s);
        }
        float Z = 0.f, a0 = 0.f, a1 = 0.f;
        for (int k = 0; k < 1024; ++k) {
            float s = (q0 * sK[2*k] + q1 * sK[2*k+1]) * rs;
            float p = expf(s - mx);
            Z  += p;
            a0 += p * sV[2*k];
            a1 += p * sV[2*k+1];
        }
        float inv = 1.f / Z;
        avbuf[(size_t)b * 4096 + f*4 + h*2 + 0] = a0 * inv;
        avbuf[(size_t)b * 4096 + f*4 + h*2 + 1] = a1 * inv;
    }
}

// ----------------------------------------- WMMA GEMM: C[32,N] = A[32,K]*Bw^T + bias
// fp32 WMMA 16x16x4; one wave per 16x16 C tile; 2 x N/16 waves total.
// A fragment layout (16x4): lane = m(0..15) + 16*kh, VGPRs hold K = 2*kh, 2*kh+1
// B fragment layout (4x16): lane = n(0..15) + 16*kh, same K split
__global__ void gemm32_wmma_kernel(const float* __restrict__ A,
                                   const float* __restrict__ Bw,   // [N,K] row-major
                                   const float* __restrict__ bias, // [N]
                                   float* __restrict__ C,          // [32,N]
                                   int K, int N)
{
    int lane = threadIdx.x & 31;
    int wid  = threadIdx.x >> 5;
    int gw   = blockIdx.x * (blockDim.x >> 5) + wid;
    int ntiles = N >> 4;
    int mt = gw / ntiles;
    int nt = gw - mt * ntiles;
    int mn = lane & 15;             // row of A tile / col of B tile
    int kk = (lane >> 4) << 1;      // K sub-offset {0,2}
    const float* ap = A  + (size_t)(mt * 16 + mn) * K + kk;
    const float* bp = Bw + (size_t)(nt * 16 + mn) * K + kk;
    __builtin_prefetch(ap, 0, 1);   // global_prefetch_b8
    __builtin_prefetch(bp, 0, 1);
    v8f acc = {};
    for (int k = 0; k < K; k += 4) {
        v2f fa, fb;
        fa.x = ap[k];     fa.y = ap[k + 1];
        fb.x = bp[k];     fb.y = bp[k + 1];
        acc = __builtin_amdgcn_wmma_f32_16x16x4_f32(false, fa, false, fb,
                                                    (short)0, acc, false, false);
    }
    int colbase = nt * 16 + (lane & 15);
    float bcol = bias[colbase];
    #pragma unroll
    for (int j = 0; j < 8; ++j) {
        int row = mt * 16 + j + ((lane >> 4) << 3);
        C[(size_t)row * N + colbase] = acc[j] + bcol;
    }
}

// -------------------------------- LayerNorm + ELU + residual, per batch row
__global__ void ln_elu_res_kernel(const float* __restrict__ hin,
                                  const float* __restrict__ t,
                                  const float* __restrict__ g,
                                  const float* __restrict__ bb,
                                  float* __restrict__ h2)
{
    __shared__ float rs[256], rs2[256];
    int b = blockIdx.x;
    const float* tr = t + (size_t)b * 1024;
    float s = 0.f, s2 = 0.f;
    for (int i = threadIdx.x; i < 1024; i += 256) { float v = tr[i]; s += v; s2 += v*v; }
    rs[threadIdx.x] = s; rs2[threadIdx.x] = s2;
    __syncthreads();
    for (int off = 128; off > 0; off >>= 1) {
        if (threadIdx.x < off) {
            rs[threadIdx.x]  += rs[threadIdx.x + off];
            rs2[threadIdx.x] += rs2[threadIdx.x + off];
        }
        __syncthreads();
    }
    float mean = rs[0] * (1.f / 1024.f);
    float var  = rs2[0] * (1.f / 1024.f) - mean * mean;
    float inv  = rsqrtf(var + BN_EPS);
    for (int i = threadIdx.x; i < 1024; i += 256) {
        float y = (tr[i] - mean) * inv * g[i] + bb[i];
        float e = y > 0.f ? y : (expf(y) - 1.f);     // ELU alpha=1
        h2[(size_t)b * 1024 + i] = hin[(size_t)b * 1024 + i] + e;
    }
}

// ----------------------------------------- motion-weight gates: mw / sum(mw)
__global__ void gates_kernel(const float* __restrict__ h2,
                             const float* __restrict__ w,     // [48,1024]
                             const float* __restrict__ bias,  // [48]
                             float* __restrict__ gates)       // [32,48]
{
    __shared__ float mw[48];
    __shared__ float sinv;
    int b = blockIdx.x, o = threadIdx.x;
    if (o < 48) {
        const float* hr = h2 + (size_t)b * 1024;
        const float* wr = w + (size_t)o * 1024;
        float acc = bias[o];
        for (int j = 0; j < 1024; ++j) acc += hr[j] * wr[j];
        mw[o] = acc;
    }
    __syncthreads();
    if (o == 0) { float s = 0.f; for (int e = 0; e < 48; ++e) s += mw[e]; sinv = 1.f / s; }
    __syncthreads();
    if (o < 48) gates[b * 48 + o] = mw[o] * sinv;
}

// ---------------------------------- grouped-expert layer, gated sum over 48
__global__ void moe_layer_kernel(const float* __restrict__ Sin,   // [32,in]
                                 const float* __restrict__ W,     // [48,out,in]
                                 const float* __restrict__ Bias,  // [48,out]
                                 const float* __restrict__ gates, // [32,48]
                                 float* __restrict__ Sout,        // [32,out]
                                 int in_dim, int out_dim)
{
    __shared__ float sin_s[32];
    __shared__ float g_s[48];
    int b = blockIdx.x, o = threadIdx.x;
    if (o < in_dim) sin_s[o] = Sin[(size_t)b * in_dim + o];
    if (o < 48)     g_s[o]   = gates[b * 48 + o];
    __syncthreads();
    if (o < out_dim) {
        float acc = 0.f;
        for (int e = 0; e < 48; ++e) {
            const float* wr = W + ((size_t)e * out_dim + o) * in_dim;
            float y = Bias[e * out_dim + o];
            for (int i = 0; i < in_dim; ++i) y += wr[i] * sin_s[i];
            acc += y * g_s[e];
        }
        Sout[(size_t)b * out_dim + o] = acc;
    }
}

// ------------------------------------------- batchnorm1d stats (B=32 small)
__global__ void bn1d_stats_kernel(const float* __restrict__ X,  // [32,out]
                                  const float* __restrict__ g,
                                  const float* __restrict__ beta,
                                  float* __restrict__ scale,
                                  float* __restrict__ shift,
                                  int out_dim)
{
    int o = threadIdx.x;
    if (o >= out_dim) return;
    float s = 0.f, s2 = 0.f;
    for (int b = 0; b < 32; ++b) {
        float v = X[(size_t)b * out_dim + o];
        s += v; s2 += v * v;
    }
    float mean = s * (1.f / 32.f);
    float var  = s2 * (1.f / 32.f) - mean * mean;
    float sc   = g[o] * rsqrtf(var + BN_EPS);
    scale[o] = sc;
    shift[o] = beta[o] - mean * sc;
}

__global__ void bn1d_apply_prelu_kernel(const float* __restrict__ X,
                                        const float* __restrict__ sc,
                                        const float* __restrict__ sh,
                                        const float* __restrict__ alpha,
                                        float* __restrict__ Y,
                                        int out_dim, int n)
{
    int i = blockIdx.x * blockDim.x + threadIdx.x;
    if (i >= n) return;
    int o = i % out_dim;
    float v = X[i] * sc[o] + sh[o];
    float a = *alpha;
    Y[i] = v >= 0.f ? v : a * v;
}

// ---------------------------------------------------------- final sigmoid
__global__ void final_kernel(const float* __restrict__ S,      // [32,128]
                             const float* __restrict__ ow,     // [128]
                             const float* __restrict__ ob,     // [1]
                             float* __restrict__ out)          // [32]
{
    int b = threadIdx.x;
    if (b >= 32) return;
    const float* sr = S + (size_t)b * 128;
    float acc = ob[0];
    for (int i = 0; i < 128; ++i) acc += sr[i] * ow[i];
    out[b] = 1.f / (1.f + expf(-acc));
}

// ======================================================================
extern "C" void kernel_launch(void* const* d_in, const int* in_sizes, int n_in,
                              void* d_out, int out_size, void* d_ws, size_t ws_size,
                              hipStream_t stream)
{
    (void)in_sizes; (void)n_in; (void)out_size; (void)ws_size;
    const float* vertex_pos = (const float*)d_in[0];
    const float* body_mesh  = (const float*)d_in[1];
    const float* conv1_w = (const float*)d_in[2];
    const float* conv1_b = (const float*)d_in[3];
    const float* bn1_g   = (const float*)d_in[4];
    const float* bn1_b   = (const float*)d_in[5];
    const float* conv2_w = (const float*)d_in[6];
    const float* conv2_b = (const float*)d_in[7];
    const float* bn2_g   = (const float*)d_in[8];
    const float* bn2_b   = (const float*)d_in[9];
    const float* prelu_a = (const float*)d_in[10];
    const float* wq = (const float*)d_in[11]; const float* bq = (const float*)d_in[12];
    const float* wk = (const float*)d_in[13]; const float* bk = (const float*)d_in[14];
    const float* wv = (const float*)d_in[15]; const float* bv = (const float*)d_in[16];
    const float* fc_w  = (const float*)d_in[17]; const float* fc_b  = (const float*)d_in[18];
    const float* ff1_w = (const float*)d_in[19]; const float* ff1_b = (const float*)d_in[20];
    const float* ln_g  = (const float*)d_in[21]; const float* ln_b  = (const float*)d_in[22];
    const float* ff2_w = (const float*)d_in[23]; const float* ff2_b = (const float*)d_in[24];
    const float* fus_w0 = (const float*)d_in[25]; const float* fus_b0 = (const float*)d_in[26];
    const float* fus_w1 = (const float*)d_in[27]; const float* fus_b1 = (const float*)d_in[28];
    const float* bnf0_g = (const float*)d_in[29]; const float* bnf0_b = (const float*)d_in[30];
    const float* bnf1_g = (const float*)d_in[31]; const float* bnf1_b = (const float*)d_in[32];
    const float* prelu_f = (const float*)d_in[33];
    const float* out_w   = (const float*)d_in[34];
    const float* out_b   = (const float*)d_in[35];

    float* ws = (float*)d_ws;
    float* X1    = ws;               // 326496
    float* X2    = X1 + 326496;      // 294912
    float* XP    = X2 + 294912;      // 294912
    float* QB    = XP + 294912;      // 131072
    float* KB    = QB + 131072;      // 131072
    float* VB    = KB + 131072;      // 131072
    float* AV    = VB + 131072;      // 131072
    float* H     = AV + 131072;      // 32768
    float* T     = H  + 32768;       // 32768
    float* H2    = T  + 32768;       // 32768
    float* GATES = H2 + 32768;       // 1536
    float* S0RAW = GATES + 1536;     // 1024
    float* S0    = S0RAW + 1024;     // 1024
    float* S1RAW = S0 + 1024;        // 4096
    float* S1    = S1RAW + 4096;     // 4096
    float* STATS = S1 + 4096;        // 384
    float* SC1 = STATS + 0,  *SH1 = STATS + 8;
    float* SC2 = STATS + 16, *SH2 = STATS + 24;
    float* SF0 = STATS + 32, *HF0 = STATS + 64;
    float* SF1 = STATS + 96, *HF1 = STATS + 224;

    conv1_kernel<<<(32*3*3401 + 255)/256, 256, 0, stream>>>(body_mesh, conv1_w, conv1_b, X1);
    bn2d_stats_kernel<<<3, 256, 0, stream>>>(X1, bn1_g, bn1_b, SC1, SH1, 3, 3401, 32);
    conv2_kernel<<<(32*3*3072 + 255)/256, 256, 0, stream>>>(X1, SC1, SH1, conv2_w, conv2_b, X2);
    bn2d_stats_kernel<<<3, 256, 0, stream>>>(X2, bn2_g, bn2_b, SC2, SH2, 3, 3072, 32);
    bn2d_apply_prelu_kernel<<<(32*9216 + 255)/256, 256, 0, stream>>>(X2, SC2, SH2, prelu_a, XP);
    qkv_kernel<<<(32*1024)/256, 256, 0, stream>>>(XP, wq, bq, wk, bk, wv, bv, QB, KB, VB);
    attn_kernel<<<64, 256, 0, stream>>>(QB, KB, VB, AV);
    // fc: [32,4096] x [4096,1024]^T  -> 128 waves, 4 waves/block
    gemm32_wmma_kernel<<<32, 128, 0, stream>>>(AV, fc_w, fc_b, H, 4096, 1024);
    // ff1: [32,1024] x [1024,1024]^T
    gemm32_wmma_kernel<<<32, 128, 0, stream>>>(H, ff1_w, ff1_b, T, 1024, 1024);
    ln_elu_res_kernel<<<32, 256, 0, stream>>>(H, T, ln_g, ln_b, H2);
    gates_kernel<<<32, 64, 0, stream>>>(H2, ff2_w, ff2_b, GATES);
    moe_layer_kernel<<<32, 64, 0, stream>>>(vertex_pos, fus_w0, fus_b0, GATES, S0RAW, 3, 32);
    bn1d_stats_kernel<<<1, 32, 0, stream>>>(S0RAW, bnf0_g, bnf0_b, SF0, HF0, 32);
    bn1d_apply_prelu_kernel<<<(32*32 + 255)/256, 256, 0, stream>>>(S0RAW, SF0, HF0, prelu_f, S0, 32, 32*32);
    moe_layer_kernel<<<32, 128, 0, stream>>>(S0, fus_w1, fus_b1, GATES, S1RAW, 32, 128);
    bn1d_stats_kernel<<<1, 128, 0, stream>>>(S1RAW, bnf1_g, bnf1_b, SF1, HF1, 128);
    bn1d_apply_prelu_kernel<<<(32*128 + 255)/256, 256, 0, stream>>>(S1RAW, SF1, HF1, prelu_f, S1, 128, 32*128);
    final_kernel<<<1, 32, 0, stream>>>(S1, out_w, out_b, (float*)d_out);
}